// SSMLayer_55336358642831
// MI455X (gfx1250) — compile-verified
//
#include <hip/hip_runtime.h>
#include <hip/hip_bf16.h>
#include <math.h>

typedef __attribute__((ext_vector_type(2))) float v2f;
typedef __attribute__((ext_vector_type(4))) float v4f;
typedef __attribute__((ext_vector_type(8))) float v8f;
typedef __attribute__((ext_vector_type(4))) int   v4i;

#define D_MODEL 512
#define D_STATE 16
#define DT_RANK 32
#define NPROJ   64          // DT_RANK + 2*D_STATE
#define BATCH   4
#define SEQLEN  2048
#define CHUNK   128
#define NCHUNK  16          // SEQLEN / CHUNK

// ---------------------------------------------------------------------------
// Async global->LDS copy (gfx1250 GLOBAL_LOAD_ASYNC_TO_LDS_B128, ASYNCcnt).
// Builtin params are int4-typed pointers: (addrspace(1) v4i*, addrspace(3)
// v4i*, imm offset, imm cpol) per the round-2 diagnostic.
// ---------------------------------------------------------------------------
__device__ __forceinline__ void async_copy_b128(const float* g, float* l) {
#if __has_builtin(__builtin_amdgcn_global_load_async_to_lds_b128)
    typedef __attribute__((address_space(1))) v4i gv4i;
    typedef __attribute__((address_space(3))) v4i lv4i;
    __builtin_amdgcn_global_load_async_to_lds_b128((gv4i*)(g), (lv4i*)(l), 0, 0);
#else
    *(v4f*)l = *(const v4f*)g;
#endif
}

__device__ __forceinline__ void async_copy_wait() {
#if __has_builtin(__builtin_amdgcn_global_load_async_to_lds_b128)
#if __has_builtin(__builtin_amdgcn_s_wait_asynccnt)
    __builtin_amdgcn_s_wait_asynccnt(0);
#else
    asm volatile("s_wait_asynccnt 0x0" ::: "memory");
#endif
#endif
}

// ---------------------------------------------------------------------------
// 1) Depthwise causal conv1d (4-tap) + bias.  idx = d + 512*(l + 2048*b)
// ---------------------------------------------------------------------------
__global__ void conv1d_kernel(const float* __restrict__ x,
                              const float* __restrict__ conv_w,
                              const float* __restrict__ conv_b,
                              float* __restrict__ xc) {
    int idx = blockIdx.x * blockDim.x + threadIdx.x;
    int d = idx & (D_MODEL - 1);
    int l = (idx >> 9) & (SEQLEN - 1);
    int b = idx >> 20;
    float w0 = conv_w[d * 4 + 0], w1 = conv_w[d * 4 + 1];
    float w2 = conv_w[d * 4 + 2], w3 = conv_w[d * 4 + 3];
    const float* xb = x + ((size_t)b * SEQLEN) * D_MODEL + d;
    float acc = conv_b[d];
    if (l >= 3) acc += xb[(size_t)(l - 3) * D_MODEL] * w0;
    if (l >= 2) acc += xb[(size_t)(l - 2) * D_MODEL] * w1;
    if (l >= 1) acc += xb[(size_t)(l - 1) * D_MODEL] * w2;
    acc += xb[(size_t)l * D_MODEL] * w3;
    xc[idx] = acc;
}

// ---------------------------------------------------------------------------
// 2) Aneg[d,n] = -exp(A_log[d,n])
// ---------------------------------------------------------------------------
__global__ void aneg_kernel(const float* __restrict__ A_log,
                            float* __restrict__ Aneg) {
    int i = blockIdx.x * blockDim.x + threadIdx.x;
    if (i < D_MODEL * D_STATE) Aneg[i] = -__expf(A_log[i]);
}

// ---------------------------------------------------------------------------
// 3) x_dbl = xc @ x_proj_w.T   (M=8192, N=64, K=512)  -- f32 WMMA 16x16x4
// ---------------------------------------------------------------------------
__global__ void xproj_wmma_kernel(const float* __restrict__ xc,
                                  const float* __restrict__ Wp,   // (64,512)
                                  float* __restrict__ xdbl) {
    const int lane = threadIdx.x & 31;
    const int wave = threadIdx.x >> 5;
    const int tile = blockIdx.x * (blockDim.x >> 5) + wave;
    const int mt = tile >> 2;            // 512 m-tiles
    const int nt = tile & 3;             // 4 n-tiles
    const int m = mt * 16 + (lane & 15);
    const int n = nt * 16 + (lane & 15);
    const int khalf = (lane >> 4) * 2;   // lanes 0-15 -> k0,k0+1 ; 16-31 -> k0+2,k0+3
    const float* arow = xc + (size_t)m * D_MODEL + khalf;
    const float* brow = Wp + (size_t)n * D_MODEL + khalf;
    v8f c = {};
    for (int k0 = 0; k0 < D_MODEL; k0 += 4) {
        v2f a = *(const v2f*)(arow + k0);
        v2f bf = *(const v2f*)(brow + k0);
        c = __builtin_amdgcn_wmma_f32_16x16x4_f32(false, a, false, bf,
                                                  (short)0, c, false, false);
    }
    const int mbase = mt * 16 + ((lane >> 4) << 3);   // lanes 16-31 hold M=r+8
    const int ncol = nt * 16 + (lane & 15);
#pragma unroll
    for (int r = 0; r < 8; ++r)
        xdbl[(size_t)(mbase + r) * NPROJ + ncol] = c[r];
}

// ---------------------------------------------------------------------------
// 4) dt = softplus(x_dbl[:, :32] @ dt_proj_w.T + b)  (M=8192, N=512, K=32)
// ---------------------------------------------------------------------------
__global__ void dtproj_wmma_kernel(const float* __restrict__ xdbl,
                                   const float* __restrict__ Wdt,  // (512,32)
                                   const float* __restrict__ bdt,  // (512,)
                                   float* __restrict__ dt) {
    const int lane = threadIdx.x & 31;
    const int wave = threadIdx.x >> 5;
    const int tile = blockIdx.x * (blockDim.x >> 5) + wave;
    const int mt = tile >> 5;            // 512 m-tiles
    const int nt = tile & 31;            // 32 n-tiles
    const int m = mt * 16 + (lane & 15);
    const int n = nt * 16 + (lane & 15);
    const int khalf = (lane >> 4) * 2;
    const float* arow = xdbl + (size_t)m * NPROJ + khalf;
    const float* brow = Wdt + (size_t)n * DT_RANK + khalf;
    v8f c = {};
#pragma unroll
    for (int k0 = 0; k0 < DT_RANK; k0 += 4) {
        v2f a = *(const v2f*)(arow + k0);
        v2f bf = *(const v2f*)(brow + k0);
        c = __builtin_amdgcn_wmma_f32_16x16x4_f32(false, a, false, bf,
                                                  (short)0, c, false, false);
    }
    const int mbase = mt * 16 + ((lane >> 4) << 3);
    const int ncol = nt * 16 + (lane & 15);
    const float bias = bdt[ncol];
#pragma unroll
    for (int r = 0; r < 8; ++r) {
        float v = c[r] + bias;
        float sp = (v > 20.0f) ? v : log1pf(__expf(v));
        dt[(size_t)(mbase + r) * D_MODEL + ncol] = sp;
    }
}

// ---------------------------------------------------------------------------
// 5) Scan pass 1: per-chunk local recurrence with h0=0.
//    Bm rows for the chunk are block-uniform -> async-staged into LDS once.
//    idx = d + 512*(chunk + 16*b); 2 blocks of 256 threads per (b,chunk).
// ---------------------------------------------------------------------------
__global__ void scan_chunk_kernel(const float* __restrict__ dt,
                                  const float* __restrict__ xc,
                                  const float* __restrict__ xdbl,
                                  const float* __restrict__ Aneg,
                                  float* __restrict__ Pbuf,
                                  float* __restrict__ Hend) {
    __shared__ __align__(16) float smem[CHUNK * D_STATE];   // 8 KB: Bm slab
    const int idx = blockIdx.x * blockDim.x + threadIdx.x;
    const int d = idx & (D_MODEL - 1);
    const int grp = blockIdx.x >> 1;             // (c + 16*b)
    const int c = grp & (NCHUNK - 1);
    const int b = grp >> 4;
    const size_t rowbase = (size_t)b * SEQLEN + (size_t)c * CHUNK;

    // Stage Bm[rowbase..rowbase+127][0..15] into LDS (async b128 per 4 floats).
#pragma unroll
    for (int it = 0; it < 2; ++it) {
        int i = threadIdx.x + 256 * it;          // 512 segments of 16B
        int r = i >> 2, s = (i & 3) * 4;
        async_copy_b128(xdbl + (rowbase + r) * NPROJ + DT_RANK + s,
                        smem + r * D_STATE + s);
    }
    async_copy_wait();
    __syncthreads();

    float An[D_STATE], h[D_STATE], P[D_STATE];
#pragma unroll
    for (int n = 0; n < D_STATE; ++n) {
        An[n] = Aneg[d * D_STATE + n];
        h[n] = 0.0f;
        P[n] = 1.0f;
    }
    for (int t = 0; t < CHUNK; ++t) {
        size_t row = rowbase + t;
        float dtv = dt[row * D_MODEL + d];
        float xv = xc[row * D_MODEL + d];
        float u = dtv * xv;
        float bmv[D_STATE];
        *(v4f*)(bmv + 0)  = *(const v4f*)(smem + t * D_STATE + 0);
        *(v4f*)(bmv + 4)  = *(const v4f*)(smem + t * D_STATE + 4);
        *(v4f*)(bmv + 8)  = *(const v4f*)(smem + t * D_STATE + 8);
        *(v4f*)(bmv + 12) = *(const v4f*)(smem + t * D_STATE + 12);
#pragma unroll
        for (int n = 0; n < D_STATE; ++n) {
            float dA = __expf(dtv * An[n]);
            h[n] = dA * h[n] + u * bmv[n];
            P[n] *= dA;
        }
    }
#pragma unroll
    for (int n = 0; n < D_STATE; ++n) {
        Pbuf[(size_t)idx * D_STATE + n] = P[n];
        Hend[(size_t)idx * D_STATE + n] = h[n];
    }
}

// ---------------------------------------------------------------------------
// 6) Combine chunk states sequentially (16 steps).  idx = n + 16*(d + 512*b)
// ---------------------------------------------------------------------------
__global__ void combine_kernel(const float* __restrict__ P,
                               const float* __restrict__ Hend,
                               float* __restrict__ Hinit) {
    int idx = blockIdx.x * blockDim.x + threadIdx.x;
    int n = idx & (D_STATE - 1);
    int d = (idx >> 4) & (D_MODEL - 1);
    int b = idx >> 13;
    float h = 0.0f;
    for (int c = 0; c < NCHUNK; ++c) {
        size_t off = (((size_t)(b * NCHUNK + c) * D_MODEL + d) << 4) + n;
        Hinit[off] = h;
        h = P[off] * h + Hend[off];
    }
}

// ---------------------------------------------------------------------------
// 7) Scan pass 2: re-run chunk recurrence with correct init state, emit
//    y = sum_n h*C + xc*D.  Bm+C slab (16 KB) async-staged into LDS.
// ---------------------------------------------------------------------------
__global__ void scan_final_kernel(const float* __restrict__ dt,
                                  const float* __restrict__ xc,
                                  const float* __restrict__ xdbl,
                                  const float* __restrict__ Aneg,
                                  const float* __restrict__ Hinit,
                                  const float* __restrict__ Dp,
                                  float* __restrict__ out) {
    __shared__ __align__(16) float smem[CHUNK * 2 * D_STATE];  // 16 KB: Bm|C slab
    const int idx = blockIdx.x * blockDim.x + threadIdx.x;
    const int d = idx & (D_MODEL - 1);
    const int grp = blockIdx.x >> 1;
    const int c = grp & (NCHUNK - 1);
    const int b = grp >> 4;
    const size_t rowbase = (size_t)b * SEQLEN + (size_t)c * CHUNK;

    // Stage [Bm|C] = xdbl cols 32..63 for the whole chunk into LDS.
#pragma unroll
    for (int it = 0; it < 4; ++it) {
        int i = threadIdx.x + 256 * it;          // 1024 segments of 16B
        int r = i >> 3, s = (i & 7) * 4;
        async_copy_b128(xdbl + (rowbase + r) * NPROJ + DT_RANK + s,
                        smem + r * 32 + s);
    }
    async_copy_wait();
    __syncthreads();

    float An[D_STATE], h[D_STATE];
#pragma unroll
    for (int n = 0; n < D_STATE; ++n) {
        An[n] = Aneg[d * D_STATE + n];
        h[n] = Hinit[(size_t)idx * D_STATE + n];
    }
    const float Dd = Dp[d];
    for (int t = 0; t < CHUNK; ++t) {
        size_t row = rowbase + t;
        float dtv = dt[row * D_MODEL + d];
        float xv = xc[row * D_MODEL + d];
        float u = dtv * xv;
        float bmv[D_STATE], cmv[D_STATE];
        *(v4f*)(bmv + 0)  = *(const v4f*)(smem + t * 32 + 0);
        *(v4f*)(bmv + 4)  = *(const v4f*)(smem + t * 32 + 4);
        *(v4f*)(bmv + 8)  = *(const v4f*)(smem + t * 32 + 8);
        *(v4f*)(bmv + 12) = *(const v4f*)(smem + t * 32 + 12);
        *(v4f*)(cmv + 0)  = *(const v4f*)(smem + t * 32 + 16);
        *(v4f*)(cmv + 4)  = *(const v4f*)(smem + t * 32 + 20);
        *(v4f*)(cmv + 8)  = *(const v4f*)(smem + t * 32 + 24);
        *(v4f*)(cmv + 12) = *(const v4f*)(smem + t * 32 + 28);
        float y = 0.0f;
#pragma unroll
        for (int n = 0; n < D_STATE; ++n) {
            float dA = __expf(dtv * An[n]);
            h[n] = dA * h[n] + u * bmv[n];
            y += h[n] * cmv[n];
        }
        out[row * D_MODEL + d] = y + xv * Dd;
    }
}

// ---------------------------------------------------------------------------
extern "C" void kernel_launch(void* const* d_in, const int* in_sizes, int n_in,
                              void* d_out, int out_size, void* d_ws, size_t ws_size,
                              hipStream_t stream) {
    const float* x         = (const float*)d_in[0];
    const float* A_log     = (const float*)d_in[1];
    const float* D         = (const float*)d_in[2];
    const float* x_proj_w  = (const float*)d_in[3];
    const float* dt_proj_w = (const float*)d_in[4];
    const float* dt_proj_b = (const float*)d_in[5];
    const float* conv_w    = (const float*)d_in[6];
    const float* conv_b    = (const float*)d_in[7];
    float* out = (float*)d_out;

    float* ws    = (float*)d_ws;
    float* xc    = ws;                       // 4,194,304
    float* xdbl  = xc + 4194304;             //   524,288
    float* dtb   = xdbl + 524288;            // 4,194,304
    float* P     = dtb + 4194304;            //   524,288
    float* Hend  = P + 524288;               //   524,288
    float* Hinit = Hend + 524288;            //   524,288
    float* Aneg  = Hinit + 524288;           //     8,192

    const int elems = BATCH * SEQLEN * D_MODEL;          // 4,194,304
    conv1d_kernel<<<elems / 256, 256, 0, stream>>>(x, conv_w, conv_b, xc);
    aneg_kernel<<<(D_MODEL * D_STATE) / 256, 256, 0, stream>>>(A_log, Aneg);

    // x_proj GEMM: 512*4 = 2048 tiles, 8 waves/block
    xproj_wmma_kernel<<<2048 / 8, 256, 0, stream>>>(xc, x_proj_w, xdbl);
    // dt_proj GEMM: 512*32 = 16384 tiles
    dtproj_wmma_kernel<<<16384 / 8, 256, 0, stream>>>(xdbl, dt_proj_w, dt_proj_b, dtb);

    const int scanThreads = BATCH * NCHUNK * D_MODEL;    // 32768
    scan_chunk_kernel<<<scanThreads / 256, 256, 0, stream>>>(dtb, xc, xdbl, Aneg, P, Hend);
    combine_kernel<<<(BATCH * D_MODEL * D_STATE) / 256, 256, 0, stream>>>(P, Hend, Hinit);
    scan_final_kernel<<<scanThreads / 256, 256, 0, stream>>>(dtb, xc, xdbl, Aneg, Hinit, D, out);
}